// Masked_INR_29566554866065
// MI455X (gfx1250) — compile-verified
//
#include <hip/hip_runtime.h>
#include <hip/hip_bf16.h>
#include <math.h>
#include <stdint.h>

typedef float v2f __attribute__((ext_vector_type(2)));
typedef float v8f __attribute__((ext_vector_type(8)));

#define NPIX   65536
#define BATCH  16
#define OFF_O1 ((size_t)0)
#define OFF_O2 ((size_t)3145728)
#define OFF_O3 ((size_t)6291456)
#define OFF_WT ((size_t)9437184)
#define WT_PER_SAMPLE 3200
#define ACT_STRIDE 78   // even (8B-aligned pairs); 14*t mod 64 distinct -> no bank conflicts

__device__ __forceinline__ float gelu_f(float v) {
    return 0.5f * v * (1.0f + erff(v * 0.70710678118654752440f));
}

// ---------------------------------------------------------------------------
// Kernel 1: per-sample top-k masks -> masked weights, zero-padded to
// [Npad][Kpad] == WMMA B-operand feed layout.
// ---------------------------------------------------------------------------
__global__ __launch_bounds__(256) void mask_kernel(
    const float* __restrict__ W0, const float* __restrict__ W1,
    const float* __restrict__ W2, const float* __restrict__ W3,
    const float* __restrict__ S0, const float* __restrict__ S1,
    const float* __restrict__ S2, const float* __restrict__ S3,
    const int* __restrict__ k0, const int* __restrict__ k1,
    const int* __restrict__ k2, const int* __restrict__ k3,
    float* __restrict__ ws)
{
    __shared__ float ss[912];
    const int l = blockIdx.x & 3;
    const int b = blockIdx.x >> 2;
    const int Oar[4] = {32, 24, 16, 3};
    const int Iar[4] = {2, 38, 33, 73};
    const int Kp[4]  = {4, 40, 36, 76};
    const int Np[4]  = {32, 32, 16, 16};
    const int Lo[4]  = {0, 128, 1408, 1984};

    const float* Wp = (l == 0) ? W0 : (l == 1) ? W1 : (l == 2) ? W2 : W3;
    const float* Sp = (l == 0) ? S0 : (l == 1) ? S1 : (l == 2) ? S2 : S3;
    const int kept  = (l == 0) ? *k0 : (l == 1) ? *k1 : (l == 2) ? *k2 : *k3;

    const int O = Oar[l], I = Iar[l], n = O * I;
    const float* Sb = Sp + (size_t)b * n;
    for (int i = threadIdx.x; i < n; i += 256) ss[i] = Sb[i];
    __syncthreads();

    float* dst = ws + OFF_WT + (size_t)b * WT_PER_SAMPLE + Lo[l];
    const int tot = Np[l] * Kp[l];
    for (int e = threadIdx.x; e < tot; e += 256) {
        const int o = e / Kp[l];
        const int k = e - o * Kp[l];
        float v = 0.0f;
        if (o < O && k < I) {
            const int idx = o * I + k;
            const float si = ss[idx];
            int cnt = 0;
            for (int j = 0; j < n; ++j) {
                const float sj = ss[j];
                cnt += (sj > si) || (sj == si && j < idx);  // stable desc rank
            }
            if (cnt < kept) v = Wp[idx];
        }
        dst[e] = v;
    }
}

// ---------------------------------------------------------------------------
// Kernel 2: fused 1x1 convs: mod_in -> gelu(48ch) -> gelu(3ch) = o1
// ---------------------------------------------------------------------------
__global__ __launch_bounds__(256) void pw_kernel(
    const float* __restrict__ mod_in,
    const float* __restrict__ w1, const float* __restrict__ b1,
    const float* __restrict__ w2, const float* __restrict__ b2,
    float* __restrict__ ws)
{
    __shared__ float sw1[144], sb1[48], sw2[144], sb2[3];
    const int tid = threadIdx.x;
    for (int i = tid; i < 144; i += 256) { sw1[i] = w1[i]; sw2[i] = w2[i]; }
    if (tid < 48) sb1[tid] = b1[tid];
    if (tid < 3)  sb2[tid] = b2[tid];
    __syncthreads();

    const size_t gid = (size_t)blockIdx.x * 256 + tid;
    const int b  = (int)(gid >> 16);
    const int nn = (int)(gid & 65535);

    float m[3];
    #pragma unroll
    for (int c = 0; c < 3; ++c) m[c] = mod_in[((size_t)b * 3 + c) * NPIX + nn];

    float o0[48];
    #pragma unroll
    for (int c = 0; c < 48; ++c)
        o0[c] = gelu_f(sw1[c*3] * m[0] + sw1[c*3+1] * m[1] + sw1[c*3+2] * m[2] + sb1[c]);

    #pragma unroll
    for (int c = 0; c < 3; ++c) {
        float a = sb2[c];
        #pragma unroll
        for (int k = 0; k < 48; ++k) a += sw2[c*48 + k] * o0[k];
        ws[OFF_O1 + ((size_t)b * 3 + c) * NPIX + nn] = gelu_f(a);
    }
}

// ---------------------------------------------------------------------------
// Kernel 3: 3x3 conv (pad=1) + residual + gelu
// ---------------------------------------------------------------------------
__global__ __launch_bounds__(256) void c3_kernel(
    const float* __restrict__ in, const float* __restrict__ w,
    const float* __restrict__ bias, float* __restrict__ outp)
{
    __shared__ float sw[81], sb[3];
    const int tid = threadIdx.x;
    if (tid < 81) sw[tid] = w[tid];
    if (tid < 3)  sb[tid] = bias[tid];
    __syncthreads();

    const size_t gid = (size_t)blockIdx.x * 256 + tid;
    const int b  = (int)(gid >> 16);
    const int nn = (int)(gid & 65535);
    const int y  = nn >> 8;
    const int xq = nn & 255;

    float inv[3][3][3];
    #pragma unroll
    for (int ic = 0; ic < 3; ++ic)
        #pragma unroll
        for (int dy = 0; dy < 3; ++dy) {
            const int yy = y + dy - 1;
            #pragma unroll
            for (int dx = 0; dx < 3; ++dx) {
                const int xx = xq + dx - 1;
                inv[ic][dy][dx] = (yy >= 0 && yy < 256 && xx >= 0 && xx < 256)
                    ? in[((size_t)b * 3 + ic) * NPIX + (size_t)yy * 256 + xx] : 0.0f;
            }
        }

    #pragma unroll
    for (int oc = 0; oc < 3; ++oc) {
        float a = sb[oc] + inv[oc][1][1];
        #pragma unroll
        for (int ic = 0; ic < 3; ++ic)
            #pragma unroll
            for (int dy = 0; dy < 3; ++dy)
                #pragma unroll
                for (int dx = 0; dx < 3; ++dx)
                    a += sw[oc*27 + ic*9 + dy*3 + dx] * inv[ic][dy][dx];
        outp[((size_t)b * 3 + oc) * NPIX + nn] = gelu_f(a);
    }
}

// ---------------------------------------------------------------------------
// Kernel 4: masked INR chain with V_WMMA_F32_16X16X4_F32 (reg-resident B)
// ---------------------------------------------------------------------------
template <int KSTEPS>
__device__ __forceinline__ v8f wmma_chainW(const float* actRow, const v2f* w, int klo)
{
    v8f c = {0.f, 0.f, 0.f, 0.f, 0.f, 0.f, 0.f, 0.f};
    #pragma unroll
    for (int ks = 0; ks < KSTEPS; ++ks) {
        v2f a;
        a[0] = actRow[ks*4 + klo];
        a[1] = actRow[ks*4 + klo + 1];
        c = __builtin_amdgcn_wmma_f32_16x16x4_f32(false, a, false, w[ks], (short)0, c, false, false);
    }
    return c;
}

__device__ __forceinline__ void store_tile(float* act, v8f c, int colBase, int lane)
{
    const int mb = (lane < 16) ? 0 : 8;
    const int ncol = colBase + (lane & 15);
    #pragma unroll
    for (int j = 0; j < 8; ++j) act[(mb + j) * ACT_STRIDE + ncol] = c[j];
}

__global__ __launch_bounds__(256) void final_kernel(
    const float* __restrict__ x, const float* __restrict__ mod_in,
    const float* __restrict__ c11w, const float* __restrict__ c11b,
    const float* __restrict__ ws, float* __restrict__ out)
{
    __shared__ __align__(16) float sWT[WT_PER_SAMPLE];  // 16B align for async b128 DMA
    __shared__ float sC[144];
    __shared__ float sB[48];
    __shared__ __align__(16) float sAct[8 * 16 * ACT_STRIDE];

    const int b     = blockIdx.x >> 5;   // 32 chunks per sample
    const int chunk = blockIdx.x & 31;
    const int tid   = threadIdx.x;

    // Async DMA the per-sample masked weights into LDS (ASYNCcnt-tracked,
    // bypasses VGPRs). 800 x b128 transfers cover 3200 floats exactly.
    {
        const float* gsrc = ws + OFF_WT + (size_t)b * WT_PER_SAMPLE;
        for (int i = tid * 4; i < WT_PER_SAMPLE; i += 1024) {
            unsigned ldsOff = (unsigned)(uintptr_t)(const void*)&sWT[i];
            asm volatile("global_load_async_to_lds_b128 %0, %1, off"
                         :: "v"(ldsOff), "v"(gsrc + i) : "memory");
        }
        asm volatile("s_wait_asynccnt 0" ::: "memory");
    }
    for (int i = tid; i < 144; i += 256) sC[i] = c11w[i];
    if (tid < 48) sB[tid] = c11b[tid];
    __syncthreads();

    const int wv   = tid >> 5;
    const int lane = tid & 31;
    const int tok  = lane & 15;
    const int half = lane >> 4;
    const int klo  = half * 2;
    float* act = &sAct[wv * 16 * ACT_STRIDE];
    float* row = &act[tok * ACT_STRIDE];     // this lane's token row
    const float* actRow = row;
    const size_t mchan = (size_t)b * 3 * NPIX;

    // Hoist all WMMA B-operands (per-lane weight pairs) into registers.
    v2f wl0[2], wl1[20], wl2[9], wl3[19];
    #pragma unroll
    for (int nt = 0; nt < 2; ++nt) {
        const float* p = &sWT[(nt*16 + tok)*4 + klo];
        wl0[nt][0] = p[0]; wl0[nt][1] = p[1];
    }
    #pragma unroll
    for (int nt = 0; nt < 2; ++nt)
        #pragma unroll
        for (int ks = 0; ks < 10; ++ks) {
            const float* p = &sWT[128 + (nt*16 + tok)*40 + ks*4 + klo];
            wl1[nt*10 + ks][0] = p[0]; wl1[nt*10 + ks][1] = p[1];
        }
    #pragma unroll
    for (int ks = 0; ks < 9; ++ks) {
        const float* p = &sWT[1408 + tok*36 + ks*4 + klo];
        wl2[ks][0] = p[0]; wl2[ks][1] = p[1];
    }
    #pragma unroll
    for (int ks = 0; ks < 19; ++ks) {
        const float* p = &sWT[1984 + tok*76 + ks*4 + klo];
        wl3[ks][0] = p[0]; wl3[ks][1] = p[1];
    }

    // cols 73..75 are never written inside the loop -> zero once
    row[73 + half] = 0.0f;
    if (half == 0) row[75] = 0.0f;

    for (int t = wv; t < 128; t += 8) {
        const int n0 = chunk * 2048 + t * 16;

        // ---- layer 0 input: x (cols 0..1), zero pad (cols 2..3) ----
        const size_t xb = ((size_t)b * NPIX + n0) * 2;
        if (half == 0) { row[0] = x[xb + tok*2 + 0]; row[1] = x[xb + tok*2 + 1]; }
        else           { row[2] = 0.0f; row[3] = 0.0f; }
        v8f h0a = wmma_chainW<1>(actRow, &wl0[0], klo);
        v8f h0b = wmma_chainW<1>(actRow, &wl0[1], klo);
        store_tile(act, h0a, 0, lane);
        store_tile(act, h0b, 16, lane);

        // ---- m0 = [o3,o2] cols 32..37; zero 38..39 (no div, coalesced) ----
        #pragma unroll
        for (int j = 0; j < 3; ++j) {
            const int ch = j*2 + half;                       // 0..5
            const size_t src = (ch < 3) ? (OFF_O3 + (size_t)ch * NPIX)
                                        : (OFF_O2 + (size_t)(ch - 3) * NPIX);
            row[32 + ch] = ws[src + mchan + n0 + tok];
        }
        row[38 + half] = 0.0f;

        v8f h1a = wmma_chainW<10>(actRow, &wl1[0],  klo);
        v8f h1b = wmma_chainW<10>(actRow, &wl1[10], klo);
        store_tile(act, h1a, 0, lane);
        store_tile(act, h1b, 16, lane);

        // ---- m1 = [o3,o2,o1] cols 24..32 (keep values for layer-3 reuse) ----
        float mv[5];
        #pragma unroll
        for (int j = 0; j < 5; ++j) {
            const int ch = j*2 + half;                       // 0..9
            if (ch < 9) {
                const size_t src = (ch < 3) ? (OFF_O3 + (size_t)ch * NPIX)
                               : (ch < 6)  ? (OFF_O2 + (size_t)(ch - 3) * NPIX)
                                           : (OFF_O1 + (size_t)(ch - 6) * NPIX);
                mv[j] = ws[src + mchan + n0 + tok];
                row[24 + ch] = mv[j];
            } else mv[j] = 0.0f;
        }
        row[33 + half] = 0.0f;
        if (half == 0) row[35] = 0.0f;

        v8f h2 = wmma_chainW<9>(actRow, wl2, klo);
        store_tile(act, h2, 0, lane);

        // ---- m2: cols 16..24 reuse mv; o0 recomputed into cols 25..72 ----
        #pragma unroll
        for (int j = 0; j < 5; ++j) {
            const int ch = j*2 + half;
            if (ch < 9) row[16 + ch] = mv[j];
        }
        const float mi0 = mod_in[mchan + 0*NPIX + n0 + tok];
        const float mi1 = mod_in[mchan + 1*NPIX + n0 + tok];
        const float mi2 = mod_in[mchan + 2*NPIX + n0 + tok];
        #pragma unroll
        for (int j = 0; j < 24; ++j) {
            const int ch = half * 24 + j;                    // halves split 48 ch
            const float v = sC[ch*3]*mi0 + sC[ch*3+1]*mi1 + sC[ch*3+2]*mi2 + sB[ch];
            row[25 + ch] = gelu_f(v);
        }

        v8f o = wmma_chainW<19>(actRow, wl3, klo);

        // ---- out[b, n0+m, ch<3]  (D: N across lanes, M across VGPRs) ----
        if (tok < 3) {
            const int mb = half * 8;
            #pragma unroll
            for (int j = 0; j < 8; ++j)
                out[((size_t)b * NPIX + n0 + mb + j) * 3 + tok] = o[j];
        }
    }
}

// ---------------------------------------------------------------------------
extern "C" void kernel_launch(void* const* d_in, const int* in_sizes, int n_in,
                              void* d_out, int out_size, void* d_ws, size_t ws_size,
                              hipStream_t stream)
{
    (void)in_sizes; (void)n_in; (void)out_size; (void)ws_size;
    const float* x      = (const float*)d_in[0];
    const float* mod_in = (const float*)d_in[1];
    const float* c11_w  = (const float*)d_in[2];
    const float* c11_b  = (const float*)d_in[3];
    const float* c12_w  = (const float*)d_in[4];
    const float* c12_b  = (const float*)d_in[5];
    const float* c21_w  = (const float*)d_in[6];
    const float* c21_b  = (const float*)d_in[7];
    const float* c22_w  = (const float*)d_in[8];
    const float* c22_b  = (const float*)d_in[9];
    const float* W0 = (const float*)d_in[10];
    const float* W1 = (const float*)d_in[11];
    const float* W2 = (const float*)d_in[12];
    const float* W3 = (const float*)d_in[13];
    const float* S0 = (const float*)d_in[14];
    const float* S1 = (const float*)d_in[15];
    const float* S2 = (const float*)d_in[16];
    const float* S3 = (const float*)d_in[17];
    const int* k0 = (const int*)d_in[18];
    const int* k1 = (const int*)d_in[19];
    const int* k2 = (const int*)d_in[20];
    const int* k3 = (const int*)d_in[21];
    float* ws  = (float*)d_ws;
    float* out = (float*)d_out;

    mask_kernel<<<64, 256, 0, stream>>>(W0, W1, W2, W3, S0, S1, S2, S3, k0, k1, k2, k3, ws);
    pw_kernel<<<BATCH * NPIX / 256, 256, 0, stream>>>(mod_in, c11_w, c11_b, c12_w, c12_b, ws);
    c3_kernel<<<BATCH * NPIX / 256, 256, 0, stream>>>(ws + OFF_O1, c21_w, c21_b, ws + OFF_O2);
    c3_kernel<<<BATCH * NPIX / 256, 256, 0, stream>>>(ws + OFF_O2, c22_w, c22_b, ws + OFF_O3);
    final_kernel<<<BATCH * 32, 256, 0, stream>>>(x, mod_in, c11_w, c11_b, ws, out);
}